// _InvAttention_257698038061
// MI455X (gfx1250) — compile-verified
//
#include <hip/hip_runtime.h>

// ---------------------------------------------------------------------------
// Problem constants (from reference): B=8, Nq=512, Nkv=512, H=16, D=256, dh=16
// ---------------------------------------------------------------------------
#define BATCH   8
#define NQ      512
#define NKV     512
#define NH      16
#define DMODEL  256
#define NEG_INF_F (-1.0e9f)

// k-chunk sizes (sized so static LDS stays < 64 KB per block)
#define KC_ATT  32     // attention kernel k-chunk
#define KC_GEMM 32     // gemm kernel k-chunk

typedef __attribute__((ext_vector_type(2))) float v2f;
typedef __attribute__((ext_vector_type(8))) float v8f;

// D = A(16x4 f32) * B(4x16 f32) + C(16x16 f32), wave32
__device__ __forceinline__ v8f wmma_f32(v2f a, v2f b, v8f c) {
    return __builtin_amdgcn_wmma_f32_16x16x4_f32(
        /*neg_a=*/false, a, /*neg_b=*/false, b,
        /*c_mod=*/(short)0, c, /*reuse_a=*/false, /*reuse_b=*/false);
}

// ---------------------------------------------------------------------------
// Kernel 1 & 3: out[n, :] = in[n, :] @ W^T + bias      (in: rows x 256,
// W: 256x256 row-major, out: rows x 256). One 16-row tile per block,
// 8 waves, each wave owns two 16-col tiles. K swept in LDS chunks of 32.
// ---------------------------------------------------------------------------
__launch_bounds__(256)
__global__ void gemm_bias_256(const float* __restrict__ in,
                              const float* __restrict__ W,
                              const float* __restrict__ bias,
                              float* __restrict__ out) {
    __shared__ float s_in[16 * 257];               // 16 rows x 256 (+1 pad)
    __shared__ float s_w[DMODEL * (KC_GEMM + 1)];  // 256 d-rows x 32 k (+1 pad)

    const int tid   = threadIdx.x;
    const int wid   = tid >> 5;
    const int lane  = tid & 31;
    const int row   = lane & 15;                   // A row / B col / C col
    const int khalf = (lane < 16) ? 0 : 2;         // f32 WMMA K split
    const int n0    = blockIdx.x * 16;

    // stage the 16x256 input tile (coalesced float4)
    for (int i = 0; i < 4; ++i) {
        int e4 = (tid + 256 * i) * 4;              // 0..4095
        int r  = e4 >> 8, k = e4 & 255;
        const float4 v = *(const float4*)&in[(size_t)(n0 + r) * DMODEL + k];
        s_in[r * 257 + k + 0] = v.x;
        s_in[r * 257 + k + 1] = v.y;
        s_in[r * 257 + k + 2] = v.z;
        s_in[r * 257 + k + 3] = v.w;
    }

    v8f acc0 = {};
    v8f acc1 = {};
    const int d0 = (wid)     * 16;                 // first  col tile
    const int d1 = (wid + 8) * 16;                 // second col tile

    for (int kc = 0; kc < DMODEL; kc += KC_GEMM) {
        __syncthreads();                           // protect s_w reuse (+ s_in on iter 0)
        // stage W[:, kc:kc+32] (coalesced float4 along each W row)
        for (int i = 0; i < 8; ++i) {
            int e4 = (tid + 256 * i) * 4;          // 0..8191
            int d  = e4 >> 5, k = e4 & 31;
            const float4 v = *(const float4*)&W[(size_t)d * DMODEL + kc + k];
            s_w[d * (KC_GEMM + 1) + k + 0] = v.x;
            s_w[d * (KC_GEMM + 1) + k + 1] = v.y;
            s_w[d * (KC_GEMM + 1) + k + 2] = v.z;
            s_w[d * (KC_GEMM + 1) + k + 3] = v.w;
        }
        __syncthreads();

        for (int ks = 0; ks < KC_GEMM; ks += 4) {
            const float* ap = &s_in[row * 257 + kc + ks + khalf];
            v2f a; a.x = ap[0]; a.y = ap[1];

            const float* bp0 = &s_w[(d0 + row) * (KC_GEMM + 1) + ks + khalf];
            v2f b0; b0.x = bp0[0]; b0.y = bp0[1];
            const float* bp1 = &s_w[(d1 + row) * (KC_GEMM + 1) + ks + khalf];
            v2f b1; b1.x = bp1[0]; b1.y = bp1[1];

            acc0 = wmma_f32(a, b0, acc0);
            acc1 = wmma_f32(a, b1, acc1);
        }
    }

    const float bias0 = bias[d0 + row];
    const float bias1 = bias[d1 + row];
    for (int i = 0; i < 8; ++i) {
        const int r = i + ((lane < 16) ? 0 : 8);
        out[(size_t)(n0 + r) * DMODEL + d0 + row] = acc0[i] + bias0;
        out[(size_t)(n0 + r) * DMODEL + d1 + row] = acc1[i] + bias1;
    }
}

// ---------------------------------------------------------------------------
// Kernel 2: fused masked softmax (online, flash-style) + per-head WMMA GEMM
// + L2-norm rescale.  Block = (b, 16-query tile), 8 waves, wave w owns heads
// 2w and 2w+1.  messages streamed once, coalesced, through LDS; next chunk
// pulled ahead with global_prefetch_b8.
// ---------------------------------------------------------------------------
#define MSG_ROWSTRIDE (KC_ATT + 1)                  // 33 floats per (h,q) row
#define MSG_HSTRIDE   (16 * MSG_ROWSTRIDE + 4)      // 532: head block (+pad)

__launch_bounds__(256)
__global__ void attn_kernel(const float* __restrict__ messages,
                            const int*   __restrict__ adj,
                            const float* __restrict__ proj,
                            float*       __restrict__ attn_out) {
    __shared__ float s_msg[NH * MSG_HSTRIDE];       // masked logits, [h][q][k]
    __shared__ float s_mask[16 * MSG_ROWSTRIDE];    // additive mask, [q][k]
    __shared__ float s_scale[8 * 32];               // per (wave, head, row)
    __shared__ float s_m[8 * 32];                   // per (wave, head, row)

    const int tid   = threadIdx.x;
    const int wid   = tid >> 5;
    const int lane  = tid & 31;
    const int row   = lane & 15;
    const int khalf = (lane < 16) ? 0 : 2;
    const int h0    = wid * 2;
    const int h1    = wid * 2 + 1;
    const int b     = blockIdx.x / (NQ / 16);
    const int q0    = (blockIdx.x % (NQ / 16)) * 16;
    const int own   = wid * 32 + lane;              // this lane's (head,row) slot

    const float* msg_base = messages + (size_t)(b * NQ + q0) * NKV * NH;
    const int*   adj_base = adj      + (size_t)(b * NQ + q0) * NKV;
    const float* proj_b   = proj     + (size_t)b * NKV * DMODEL;

    float m = -3.0e38f, l = 0.0f, s2 = 0.0f;        // this lane's row stats
    v8f acc0 = {};                                  // head h0 accumulator
    v8f acc1 = {};                                  // head h1 accumulator

    for (int kc = 0; kc < NKV; kc += KC_ATT) {
        __syncthreads();                            // s_msg/s_mask safe to rewrite
        // ---- additive mask for this (qtile, kchunk): 16x32 ints, coalesced
        for (int i = 0; i < 2; ++i) {
            int e = tid + 256 * i;                  // 0..511
            int q = e >> 5, k = e & 31;
            s_mask[q * MSG_ROWSTRIDE + k] =
                (adj_base[(size_t)q * NKV + kc + k] > 0) ? 0.0f : NEG_INF_F;
        }
        __syncthreads();
        // ---- masked logits: 16q x 32k x 16h floats, float4-coalesced along h
        for (int i = 0; i < 8; ++i) {
            int e4 = (tid + 256 * i) * 4;           // 0..8191
            int q  = e4 >> 9;
            int j  = e4 & 511;                      // k*16 + h, h % 4 == 0
            int k  = j >> 4, h = j & 15;
            const float4 v =
                *(const float4*)&msg_base[(size_t)q * (NKV * NH) + kc * NH + j];
            const float mk = s_mask[q * MSG_ROWSTRIDE + k];
            s_msg[(h + 0) * MSG_HSTRIDE + q * MSG_ROWSTRIDE + k] = v.x + mk;
            s_msg[(h + 1) * MSG_HSTRIDE + q * MSG_ROWSTRIDE + k] = v.y + mk;
            s_msg[(h + 2) * MSG_HSTRIDE + q * MSG_ROWSTRIDE + k] = v.z + mk;
            s_msg[(h + 3) * MSG_HSTRIDE + q * MSG_ROWSTRIDE + k] = v.w + mk;
        }

        // ---- prefetch next chunk toward L2/L0 while this chunk is consumed
        // (global_prefetch_b8: no VGPR return, no LOADcnt).  One 128B line per
        // thread covers the whole next 32KB message slab; threads 0..15 cover
        // the 16 next adjacency rows.
        if (kc + KC_ATT < NKV) {
            const int pq = tid >> 4;                // 0..15
            const int pj = (tid & 15) * 32;         // 0..480, 32 floats = 128B
            __builtin_prefetch(&msg_base[(size_t)pq * (NKV * NH) +
                                         (kc + KC_ATT) * NH + pj], 0, 3);
            if (tid < 16) {
                __builtin_prefetch(&adj_base[(size_t)tid * NKV + kc + KC_ATT], 0, 3);
            }
        }
        __syncthreads();

        // ---- online softmax stats: lanes 0-15 -> head h0, lanes 16-31 -> h1
        {
            const int myh = (lane < 16) ? h0 : h1;
            const float* rp = &s_msg[myh * MSG_HSTRIDE + row * MSG_ROWSTRIDE];
            float mc = -3.0e38f;
            for (int k = 0; k < KC_ATT; ++k) mc = fmaxf(mc, rp[k]);
            const float mnew = fmaxf(m, mc);
            const float r    = __expf(m - mnew);    // 0 on first chunk
            float lc = 0.0f, s2c = 0.0f;
            for (int k = 0; k < KC_ATT; ++k) {
                const float t = __expf(rp[k] - mnew);
                lc  += t;
                s2c += t * t;                       // exp(2*(x-m)) = t^2
            }
            l  = l * r + lc;
            s2 = s2 * (r * r) + s2c;
            m  = mnew;
            s_scale[own] = r;                       // same-wave producers/consumers
            s_m[own]     = mnew;
        }

        // ---- rescale accumulators by exp(m_old - m_new), per C row
        for (int i = 0; i < 8; ++i) {
            const int rr = i + ((lane < 16) ? 0 : 8);
            acc0[i] *= s_scale[wid * 32 + rr];
            acc1[i] *= s_scale[wid * 32 + 16 + rr];
        }
        const float m0 = s_m[wid * 32 + row];       // row max for head h0
        const float m1 = s_m[wid * 32 + 16 + row];  // row max for head h1

        // ---- WMMA: P(16 x 32) @ headfeats(32 x 16), K in slices of 4
        const float* p0 = &s_msg[h0 * MSG_HSTRIDE + row * MSG_ROWSTRIDE + khalf];
        const float* p1 = &s_msg[h1 * MSG_HSTRIDE + row * MSG_ROWSTRIDE + khalf];
        for (int ks = 0; ks < KC_ATT; ks += 4) {
            v2f a0; a0.x = __expf(p0[ks] - m0); a0.y = __expf(p0[ks + 1] - m0);
            v2f a1; a1.x = __expf(p1[ks] - m1); a1.y = __expf(p1[ks + 1] - m1);

            const float* vrow = proj_b + (size_t)(kc + ks + khalf) * DMODEL;
            v2f b0; b0.x = vrow[h0 * 16 + row]; b0.y = vrow[DMODEL + h0 * 16 + row];
            v2f b1; b1.x = vrow[h1 * 16 + row]; b1.y = vrow[DMODEL + h1 * 16 + row];

            acc0 = wmma_f32(a0, b0, acc0);
            acc1 = wmma_f32(a1, b1, acc1);
        }
    }

    // ---- final per-row factor: (attn_out/l) * (sqrt(s2)/l) = acc * sqrt(s2)/l^2
    s_scale[own] = sqrtf(s2) / (l * l);
    for (int i = 0; i < 8; ++i) {
        const int rr = i + ((lane < 16) ? 0 : 8);
        const float f0 = s_scale[wid * 32 + rr];
        const float f1 = s_scale[wid * 32 + 16 + rr];
        attn_out[(size_t)(b * NQ + q0 + rr) * DMODEL + h0 * 16 + row] = acc0[i] * f0;
        attn_out[(size_t)(b * NQ + q0 + rr) * DMODEL + h1 * 16 + row] = acc1[i] * f1;
    }
}

// ---------------------------------------------------------------------------
// Launch: proj GEMM -> fused attention -> output GEMM (stream-ordered)
// ---------------------------------------------------------------------------
extern "C" void kernel_launch(void* const* d_in, const int* in_sizes, int n_in,
                              void* d_out, int out_size, void* d_ws, size_t ws_size,
                              hipStream_t stream) {
    (void)in_sizes; (void)n_in; (void)out_size; (void)ws_size;

    const float* v_inv    = (const float*)d_in[0];
    const float* messages = (const float*)d_in[1];
    const int*   adj      = (const int*)  d_in[2];
    const float* W_in     = (const float*)d_in[3];
    const float* b_in     = (const float*)d_in[4];
    const float* W_out    = (const float*)d_in[5];
    const float* b_out    = (const float*)d_in[6];
    float*       out      = (float*)d_out;

    float* proj = (float*)d_ws;                                    // 4 MB
    float* attn = proj + (size_t)BATCH * NKV * DMODEL;             // 4 MB

    // proj = v_inv @ W_in^T + b_in        (4096 x 256 rows)
    gemm_bias_256<<<(BATCH * NKV) / 16, 256, 0, stream>>>(v_inv, W_in, b_in, proj);
    // fused masked-softmax attention + L2 rescale
    attn_kernel<<<BATCH * (NQ / 16), 256, 0, stream>>>(messages, adj, proj, attn);
    // out = attn @ W_out^T + b_out        (4096 x 256 rows)
    gemm_bias_256<<<(BATCH * NQ) / 16, 256, 0, stream>>>(attn, W_out, b_out, out);
}